// Voice_Time_CrossAttention_11742440587575
// MI455X (gfx1250) — compile-verified
//
#include <hip/hip_runtime.h>

// ---------------------------------------------------------------------------
// Voice_Time_CrossAttention on MI455X (gfx1250, wave32)
//   B=1, T=4096, E=512, H=512
// ---------------------------------------------------------------------------

#define T_LEN 4096
#define E_DIM 512
#define H_DIM 512
#define H3    1536   // 3*H
#define H2    1024   // 2*H

typedef __attribute__((ext_vector_type(2))) float v2f;
typedef __attribute__((ext_vector_type(8))) float v8f;

// ---------------------------------------------------------------------------
// Kernel 1: e[t] = dot(query[t,:], fc_w) + fc_b          (one wave per t)
// ---------------------------------------------------------------------------
__global__ __launch_bounds__(256) void e_kernel(const float* __restrict__ query,
                                                const float* __restrict__ fc_w,
                                                const float* __restrict__ fc_b,
                                                float* __restrict__ e) {
    const int lane = threadIdx.x & 31;
    const int wave = threadIdx.x >> 5;
    const int t = blockIdx.x * 8 + wave;
    const float* q = query + t * E_DIM;
    float acc = 0.f;
    #pragma unroll
    for (int j = lane; j < E_DIM; j += 32) acc += q[j] * fc_w[j];
    #pragma unroll
    for (int off = 16; off; off >>= 1) acc += __shfl_xor(acc, off, 32);
    if (lane == 0) e[t] = acc + fc_b[0];
}

// ---------------------------------------------------------------------------
// Kernel 2: write energy = diag(e) (67 MB, float4 stores) and zero lin region
//   d_out layout: [0,1024) = lin, [1024, 1024+T*T) = energy
// ---------------------------------------------------------------------------
__global__ __launch_bounds__(256) void energy_kernel(const float* __restrict__ e,
                                                     float* __restrict__ out) {
    const unsigned id = blockIdx.x * 256u + threadIdx.x;   // float4 index
    if (blockIdx.x == 0 && threadIdx.x < 256) {            // zero lin (1024 f = 256 f4)
        float4 z = make_float4(0.f, 0.f, 0.f, 0.f);
        ((float4*)out)[threadIdx.x] = z;
    }
    const unsigned base = id * 4u;                         // element index in energy
    const int row = (int)(base / T_LEN);
    const int col = (int)(base % T_LEN);
    float4 v = make_float4(0.f, 0.f, 0.f, 0.f);
    if (row >= col && row < col + 4) {
        float ev = e[row];
        int slot = row - col;
        if (slot == 0) v.x = ev; else if (slot == 1) v.y = ev;
        else if (slot == 2) v.z = ev; else v.w = ev;
    }
    ((float4*)(out + 1024))[id] = v;
}

// ---------------------------------------------------------------------------
// Kernel 3: WMMA GEMM  xp = (e .* input) @ W_ih^T + b_ih   for both directions
//   M=4096, K=512, N=1536, f32 via V_WMMA_F32_16X16X4_F32.
//   Per wave: 16x64 tile (4 accumulators, A reused). Block = 8 waves = 32x256.
//   A-frag (16x4): lane l -> M = l%16, K = 2*(l/16)+{0,1}   (contiguous float2)
//   B-frag (4x16): lane l -> N = l%16, K = 2*(l/16)+{0,1}   (contiguous float2)
//   C/D (16x16):   lane l -> N = l%16, VGPR r -> M = r + 8*(l/16)
// ---------------------------------------------------------------------------
__global__ __launch_bounds__(256) void gemm_xp_kernel(
    const float* __restrict__ input, const float* __restrict__ e,
    const float* __restrict__ w_ih_f, const float* __restrict__ b_ih_f,
    const float* __restrict__ w_ih_b, const float* __restrict__ b_ih_b,
    float* __restrict__ xp_f, float* __restrict__ xp_b) {

    const int dir = blockIdx.z;
    const float* __restrict__ W    = dir ? w_ih_b : w_ih_f;
    const float* __restrict__ bias = dir ? b_ih_b : b_ih_f;
    float* __restrict__ out        = dir ? xp_b   : xp_f;

    const int tid  = threadIdx.x;
    const int wave = tid >> 5, lane = tid & 31;
    const int m0 = blockIdx.y * 32 + (wave & 1) * 16;
    const int n0 = blockIdx.x * 256 + (wave >> 1) * 64;
    const int row = lane & 15;            // M for A-frag; N for B/C/D frags
    const int kq  = (lane >> 4) << 1;     // 0 or 2

    const float escale = e[m0 + row];     // fold x = e[m]*input[m,:] into A load
    const v2f* __restrict__ Arow = (const v2f*)(input + (size_t)(m0 + row) * E_DIM);

    v8f acc[4];
    #pragma unroll
    for (int s = 0; s < 4; ++s) {
        const float bv = bias[n0 + s * 16 + row];
        #pragma unroll
        for (int r = 0; r < 8; ++r) acc[s][r] = bv;
    }

    for (int k = 0; k < E_DIM; k += 4) {
        const int kp = (k + kq) >> 1;
        v2f a = Arow[kp] * escale;
        #pragma unroll
        for (int s = 0; s < 4; ++s) {
            const v2f* __restrict__ Brow =
                (const v2f*)(W + (size_t)(n0 + s * 16 + row) * E_DIM);
            v2f b = Brow[kp];
            acc[s] = __builtin_amdgcn_wmma_f32_16x16x4_f32(
                false, a, false, b, (short)0, acc[s], false, false);
        }
    }

    const int mbase = m0 + ((lane >> 4) << 3);
    #pragma unroll
    for (int s = 0; s < 4; ++s) {
        const int n = n0 + s * 16 + row;
        #pragma unroll
        for (int r = 0; r < 8; ++r)
            out[(size_t)(mbase + r) * H3 + n] = acc[s][r];
    }
}

// ---------------------------------------------------------------------------
// Kernel 4: sequential GRU scan, one persistent 1024-thread WG per direction.
//   h (2KB) and hp (6KB) in LDS; W_hh (3MB/dir) streamed from L2-resident mem.
//   outputs[t, 0:512) = forward h_t ; outputs[t, 512:1024) = backward h at t.
// ---------------------------------------------------------------------------
__global__ __launch_bounds__(1024) void gru_kernel(
    const float* __restrict__ xp_f, const float* __restrict__ xp_b,
    const float* __restrict__ w_hh_f, const float* __restrict__ b_hh_f,
    const float* __restrict__ w_hh_b, const float* __restrict__ b_hh_b,
    float* __restrict__ outputs) {

    const int dir = blockIdx.x;
    const float* __restrict__ xp   = dir ? xp_b   : xp_f;
    const float* __restrict__ W    = dir ? w_hh_b : w_hh_f;
    const float* __restrict__ bias = dir ? b_hh_b : b_hh_f;

    __shared__ float h[H_DIM];
    __shared__ float hp[H3];
    const int tid = threadIdx.x;
    if (tid < H_DIM) h[tid] = 0.f;
    __syncthreads();

    for (int step = 0; step < T_LEN; ++step) {
        // phase 1: hp = h @ W_hh^T + b_hh   (rows tid, and tid+1024 if tid<512)
        {
            const float4* __restrict__ wr = (const float4*)(W + (size_t)tid * H_DIM);
            float acc = 0.f;
            #pragma unroll 8
            for (int k = 0; k < H_DIM / 4; ++k) {
                float4 w4 = wr[k];
                acc += w4.x * h[4 * k]     + w4.y * h[4 * k + 1]
                     + w4.z * h[4 * k + 2] + w4.w * h[4 * k + 3];
            }
            hp[tid] = acc + bias[tid];
        }
        if (tid < H_DIM) {
            const int j = tid + 1024;
            const float4* __restrict__ wr = (const float4*)(W + (size_t)j * H_DIM);
            float acc = 0.f;
            #pragma unroll 8
            for (int k = 0; k < H_DIM / 4; ++k) {
                float4 w4 = wr[k];
                acc += w4.x * h[4 * k]     + w4.y * h[4 * k + 1]
                     + w4.z * h[4 * k + 2] + w4.w * h[4 * k + 3];
            }
            hp[j] = acc + bias[j];
        }
        __syncthreads();
        // phase 2: gates + state update (PyTorch order r,z,n)
        if (tid < H_DIM) {
            const float* __restrict__ xpt = xp + (size_t)step * H3;
            const float xr = xpt[tid], xz = xpt[H_DIM + tid], xn = xpt[2 * H_DIM + tid];
            const float r = 1.f / (1.f + __expf(-(xr + hp[tid])));
            const float z = 1.f / (1.f + __expf(-(xz + hp[H_DIM + tid])));
            const float n = tanhf(xn + r * hp[2 * H_DIM + tid]);
            const float hn = (1.f - z) * n + z * h[tid];
            h[tid] = hn;                                  // own slot only: safe
            const int t_mem = dir ? (T_LEN - 1 - step) : step;
            outputs[(size_t)t_mem * H2 + dir * H_DIM + tid] = hn;
        }
        __syncthreads();
    }
}

// ---------------------------------------------------------------------------
// Kernel 5: scores[t] = dot(outputs[t,:], hid) / sqrt(2H)   (one wave per t)
//   hid = concat(hb_last, hf_last) = [outputs[0,512:1024), outputs[T-1,0:512)]
// ---------------------------------------------------------------------------
__global__ __launch_bounds__(256) void scores_kernel(const float* __restrict__ outputs,
                                                     float* __restrict__ scores) {
    __shared__ float hid[H2];
    const int tid = threadIdx.x;
    for (int i = tid; i < H2; i += 256)
        hid[i] = (i < H_DIM) ? outputs[H_DIM + i]
                             : outputs[(size_t)(T_LEN - 1) * H2 + (i - H_DIM)];
    __syncthreads();
    const int lane = tid & 31, wave = tid >> 5;
    const int t = blockIdx.x * 8 + wave;
    const float* __restrict__ rowp = outputs + (size_t)t * H2;
    float acc = 0.f;
    #pragma unroll
    for (int j = lane; j < H2; j += 32) acc += rowp[j] * hid[j];
    #pragma unroll
    for (int off = 16; off; off >>= 1) acc += __shfl_xor(acc, off, 32);
    if (lane == 0) scores[t] = acc * 0.03125f;   // 1/sqrt(1024)
}

// ---------------------------------------------------------------------------
// Kernel 6: att = softmax(scores) over T=4096   (single 1024-thread WG)
// ---------------------------------------------------------------------------
__global__ __launch_bounds__(1024) void softmax_kernel(const float* __restrict__ scores,
                                                       float* __restrict__ att) {
    __shared__ float red[32];
    __shared__ float bcast;
    const int tid = threadIdx.x, lane = tid & 31, wid = tid >> 5;
    float v[4];
    float m = -1e30f;
    #pragma unroll
    for (int i = 0; i < 4; ++i) { v[i] = scores[tid + i * 1024]; m = fmaxf(m, v[i]); }
    #pragma unroll
    for (int off = 16; off; off >>= 1) m = fmaxf(m, __shfl_xor(m, off, 32));
    if (lane == 0) red[wid] = m;
    __syncthreads();
    if (wid == 0) {
        float x = red[lane];
        #pragma unroll
        for (int off = 16; off; off >>= 1) x = fmaxf(x, __shfl_xor(x, off, 32));
        if (lane == 0) bcast = x;
    }
    __syncthreads();
    m = bcast;
    float ex[4], s = 0.f;
    #pragma unroll
    for (int i = 0; i < 4; ++i) { ex[i] = __expf(v[i] - m); s += ex[i]; }
    #pragma unroll
    for (int off = 16; off; off >>= 1) s += __shfl_xor(s, off, 32);
    __syncthreads();
    if (lane == 0) red[wid] = s;
    __syncthreads();
    if (wid == 0) {
        float x = red[lane];
        #pragma unroll
        for (int off = 16; off; off >>= 1) x += __shfl_xor(x, off, 32);
        if (lane == 0) bcast = x;
    }
    __syncthreads();
    const float inv = 1.f / bcast;
    #pragma unroll
    for (int i = 0; i < 4; ++i) att[tid + i * 1024] = ex[i] * inv;
}

// ---------------------------------------------------------------------------
// Kernel 7: lin[j] = sum_t att[t] * outputs[t][j]  (partials via atomicAdd;
//   lin region of d_out was zeroed by energy_kernel earlier in the stream)
// ---------------------------------------------------------------------------
__global__ __launch_bounds__(1024) void wsum_kernel(const float* __restrict__ outputs,
                                                    const float* __restrict__ att,
                                                    float* __restrict__ lin) {
    const int j = threadIdx.x;
    const int t0 = blockIdx.x * (T_LEN / 16);
    float acc = 0.f;
    for (int t = t0; t < t0 + T_LEN / 16; ++t)
        acc += att[t] * outputs[(size_t)t * H2 + j];
    atomicAdd(&lin[j], acc);
}

// ---------------------------------------------------------------------------
extern "C" void kernel_launch(void* const* d_in, const int* in_sizes, int n_in,
                              void* d_out, int out_size, void* d_ws, size_t ws_size,
                              hipStream_t stream) {
    const float* input  = (const float*)d_in[0];
    const float* query  = (const float*)d_in[1];
    const float* fc_w   = (const float*)d_in[2];
    const float* fc_b   = (const float*)d_in[3];
    const float* w_ih_f = (const float*)d_in[4];
    const float* w_hh_f = (const float*)d_in[5];
    const float* b_ih_f = (const float*)d_in[6];
    const float* b_hh_f = (const float*)d_in[7];
    const float* w_ih_b = (const float*)d_in[8];
    const float* w_hh_b = (const float*)d_in[9];
    const float* b_ih_b = (const float*)d_in[10];
    const float* b_hh_b = (const float*)d_in[11];
    float* out = (float*)d_out;

    // workspace layout (floats)
    float* ws = (float*)d_ws;
    float* e_buf   = ws;                           // 4096
    float* scores  = ws + 4096;                    // 4096
    float* att     = ws + 8192;                    // 4096
    float* xp_f    = ws + 16384;                   // 4096*1536
    float* xp_b    = xp_f + (size_t)T_LEN * H3;    // 4096*1536
    float* outputs = xp_b + (size_t)T_LEN * H3;    // 4096*1024

    e_kernel<<<T_LEN / 8, 256, 0, stream>>>(query, fc_w, fc_b, e_buf);
    energy_kernel<<<(T_LEN * (T_LEN / 4)) / 256, 256, 0, stream>>>(e_buf, out);
    gemm_xp_kernel<<<dim3(H3 / 256, T_LEN / 32, 2), 256, 0, stream>>>(
        input, e_buf, w_ih_f, b_ih_f, w_ih_b, b_ih_b, xp_f, xp_b);
    gru_kernel<<<2, 1024, 0, stream>>>(xp_f, xp_b, w_hh_f, b_hh_f, w_hh_b, b_hh_b,
                                       outputs);
    scores_kernel<<<T_LEN / 8, 256, 0, stream>>>(outputs, scores);
    softmax_kernel<<<1, 1024, 0, stream>>>(scores, att);
    wsum_kernel<<<16, 1024, 0, stream>>>(outputs, att, out);
}